// DCGRU_7206955123414
// MI455X (gfx1250) — compile-verified
//
#include <hip/hip_runtime.h>
#include <math.h>

typedef __attribute__((ext_vector_type(2))) float v2f;
typedef __attribute__((ext_vector_type(8))) float v8f;

constexpr int NN   = 64;    // nodes
constexpr int TT   = 200;   // time steps
constexpr int BB   = 64;    // batch
constexpr int HH   = 64;    // hidden
constexpr int G3   = 192;   // 3*H gates per direction
constexpr int GALL = 384;   // both directions

// workspace layout (float offsets)
constexpr size_t WS_LTIL = 0;                      // [64][64]
constexpr size_t WS_SG   = 4096;                   // [64][64]
constexpr size_t WS_L2   = 8192;                   // [64][64]
constexpr size_t WS_M    = 12288;                  // [384][64] fused weight
constexpr size_t WS_BIAS = 36864;                  // [384]
constexpr size_t WS_XP   = 37376;                  // [64][200][384] gate preacts
constexpr size_t WS_GRU  = WS_XP + (size_t)BB*TT*GALL;  // [64][200][128]

__device__ __forceinline__ v8f wmma4(v2f a, v2f b, v8f c) {
  // V_WMMA_F32_16X16X4_F32: D = A(16x4) x B(4x16) + C
  return __builtin_amdgcn_wmma_f32_16x16x4_f32(false, a, false, b, (short)0, c,
                                               false, false);
}

__device__ __forceinline__ float sigmoidf_(float v) {
  return 1.0f / (1.0f + __expf(-v));
}

// ---------------------------------------------------------------- K0: graphs
__global__ void k_build_graphs(const int* sei, const float* sew,
                               const int* fei, const float* few, float* ws) {
  __shared__ float Lt[NN * NN];
  __shared__ float Sg[NN * NN];
  __shared__ float degs[NN];
  __shared__ float degg[NN];
  int tid = threadIdx.x;
  for (int i = tid; i < NN * NN; i += blockDim.x) { Lt[i] = 0.f; Sg[i] = 0.f; }
  if (tid < NN) { degs[tid] = 0.f; degg[tid] = 0.f; }
  __syncthreads();
  for (int e = tid; e < 512; e += blockDim.x) atomicAdd(&degs[sei[e]], sew[e]);
  for (int e = tid; e < 1024; e += blockDim.x) atomicAdd(&degg[fei[1024 + e]], few[e]);
  __syncthreads();
  if (tid < NN) {
    float d = degs[tid];
    degs[tid] = (d > 0.f) ? rsqrtf(d) : 0.f;
    degg[tid] = rsqrtf(degg[tid] + 1.0f);
  }
  __syncthreads();
  for (int e = tid; e < 512; e += blockDim.x) {
    int r = sei[e], c = sei[512 + e];
    atomicAdd(&Lt[c * NN + r], -(degs[r] * sew[e] * degs[c]));  // Ltil = -S
  }
  for (int e = tid; e < 1024; e += blockDim.x) {
    int r = fei[e], c = fei[1024 + e];
    atomicAdd(&Sg[c * NN + r], degg[r] * few[e] * degg[c]);
  }
  __syncthreads();
  if (tid < NN) Sg[tid * NN + tid] += degg[tid] * degg[tid];
  __syncthreads();
  for (int i = tid; i < NN * NN; i += blockDim.x) {
    ws[WS_LTIL + i] = Lt[i];
    ws[WS_SG + i]   = Sg[i];
  }
  for (int idx = tid; idx < NN * NN; idx += blockDim.x) {   // L2 = Ltil @ Ltil
    int i = idx >> 6, j = idx & 63;
    float acc = 0.f;
    for (int k = 0; k < NN; ++k) acc += Lt[i * NN + k] * Lt[k * NN + j];
    ws[WS_L2 + idx] = acc;
  }
}

// -------------------------------------------------------- K1: fold weights
__global__ void k_fuse_weights(const float* Wcheb, const float* bcheb,
                               const float* Wgcn, const float* bgcn,
                               const float* Wih_f, const float* bih_f,
                               const float* Wih_b, const float* bih_b,
                               float* ws) {
  int g = blockIdx.x;   // 0..383 (0..191 fwd, 192..383 bwd)
  int n = threadIdx.x;  // 0..63
  const float* Wih = (g < G3) ? Wih_f : Wih_b;
  const float* bih = (g < G3) ? bih_f : bih_b;
  int gl = g % G3;
  __shared__ float A1s[NN], A2s[NN], Gs[NN], Bred[NN];
  const float* wr = Wih + (size_t)gl * (2 * HH * NN) + (size_t)n * (2 * HH);
  float a0 = 0.f, a1 = 0.f, a2 = 0.f, gg = 0.f, bp = 0.f;
  for (int h = 0; h < HH; ++h) {
    float w = wr[h];
    a0 += Wcheb[h] * w;
    a1 += Wcheb[HH + h] * w;
    a2 += Wcheb[2 * HH + h] * w;
    bp += bcheb[h] * w;
    float w2 = wr[HH + h];
    gg += Wgcn[h] * w2;
    bp += bgcn[h] * w2;
  }
  A1s[n] = a1; A2s[n] = a2; Gs[n] = gg; Bred[n] = bp;
  __syncthreads();
  const float* Lt = ws + WS_LTIL;
  const float* L2 = ws + WS_L2;
  const float* Sg = ws + WS_SG;
  float acc = a0 - a2;  // A0 - A2 (from the -I in 2L^2 - I)
  for (int k = 0; k < NN; ++k)
    acc += A1s[k] * Lt[k * NN + n] + 2.0f * A2s[k] * L2[k * NN + n] +
           Gs[k] * Sg[k * NN + n];
  ws[WS_M + (size_t)g * NN + n] = acc;
  if (n == 0) {
    float b = bih[gl];
    for (int k = 0; k < NN; ++k) b += Bred[k];
    ws[WS_BIAS + g] = b;
  }
}

// --------------------------------------------- K2: xp = M @ x + bias (WMMA)
__global__ void __launch_bounds__(256) k_xp_gemm(const float* x, float* ws) {
  __shared__ float Msh[64 * NN];
  __shared__ float bsh[64];
  int gc  = blockIdx.x;  // gate chunk 0..5 (64 gates each)
  int b   = blockIdx.y;  // batch
  int tid = threadIdx.x;
  const float* M = ws + WS_M + (size_t)gc * 64 * NN;
  for (int i = tid; i < 64 * NN; i += blockDim.x) Msh[i] = M[i];
  if (tid < 64) bsh[tid] = ws[WS_BIAS + gc * 64 + tid];
  __syncthreads();
  int wave = tid >> 5, lane = tid & 31;
  int l15  = lane & 15;
  int koff = (lane >> 4) * 2;
  const float* xb = x + (size_t)b * NN * TT;
  float* xp = ws + WS_XP + (size_t)b * TT * GALL + (size_t)gc * 64;
  for (int tile = wave; tile < 52; tile += 8) {   // 13 t-tiles x 4 gate-tiles
    int tt = tile % 13, gt = tile / 13;
    int t0  = tt * 16;
    int col = gt * 16 + l15;
    float bias = bsh[col];
    v8f c = { bias, bias, bias, bias, bias, bias, bias, bias };
    int trow = t0 + l15;
    if (trow > TT - 1) trow = TT - 1;   // clamp loads; stores are guarded
    __builtin_prefetch(xb + (size_t)koff * TT + trow, 0, 1);
    for (int ks = 0; ks < 16; ++ks) {   // K = 64 nodes, 4 per step
      int k0 = ks * 4 + koff;
      v2f a;
      a.x = xb[(size_t)k0 * TT + trow];
      a.y = xb[(size_t)(k0 + 1) * TT + trow];
      v2f bf = *(const v2f*)&Msh[col * NN + k0];
      c = wmma4(a, bf, c);
    }
    for (int i = 0; i < 8; ++i) {
      int row = i + ((lane >> 4) << 3);
      int t = t0 + row;
      if (t < TT) xp[(size_t)t * GALL + col] = c[i];
    }
  }
}

// ---------------------------------------- K3: GRU recurrence (WMMA, per dir)
__global__ void __launch_bounds__(512) k_gru(const float* Whh_f, const float* bhh_f,
                                             const float* Whh_b, const float* bhh_b,
                                             float* ws) {
  extern __shared__ float lds[];     // 112KB dynamic LDS (CDNA5: 320KB/WGP)
  float* Hs  = lds;                  // [64][64]  hidden state
  float* Wsh = lds + 4096;           // [192][64] Whh
  float* Hg  = lds + 16384;          // [64][192] H @ Whh^T + bhh
  int dir = blockIdx.x;
  const float* Whh = dir ? Whh_b : Whh_f;
  const float* bhh = dir ? bhh_b : bhh_f;
  int tid = threadIdx.x;  // 512 threads = 16 waves
  for (int i = tid; i < BB * HH; i += 512) Hs[i] = 0.f;
  for (int i = tid; i < G3 * HH; i += 512) Wsh[i] = Whh[i];
  __syncthreads();
  int wave = tid >> 5, lane = tid & 31;
  int l15  = lane & 15;
  int koff = (lane >> 4) * 2;
  const float* xpb = ws + WS_XP + (size_t)dir * G3;
  float* gru = ws + WS_GRU;
  for (int s = 0; s < TT; ++s) {
    int t = dir ? (TT - 1 - s) : s;
    // Hg[64,192] = Hs[64,64] @ Whh^T + bhh  -- 48 tiles over 16 waves
    for (int tile = wave; tile < 48; tile += 16) {
      int bt = tile & 3, gt = tile >> 2;
      int col = gt * 16 + l15;
      float bias = bhh[col];
      v8f c = { bias, bias, bias, bias, bias, bias, bias, bias };
      int row = bt * 16 + l15;
      for (int ks = 0; ks < 16; ++ks) {
        int k0 = ks * 4 + koff;
        v2f a  = *(const v2f*)&Hs[row * HH + k0];
        v2f bf = *(const v2f*)&Wsh[col * HH + k0];
        c = wmma4(a, bf, c);
      }
      for (int i = 0; i < 8; ++i) {
        int r = bt * 16 + i + ((lane >> 4) << 3);
        Hg[r * G3 + col] = c[i];
      }
    }
    __syncthreads();
    // gate elementwise + state update
    for (int idx = tid; idx < BB * HH; idx += 512) {
      int b = idx >> 6, h = idx & 63;
      const float* xv = xpb + ((size_t)b * TT + t) * GALL;
      float xr = xv[h], xz = xv[HH + h], xn = xv[2 * HH + h];
      float hr = Hg[b * G3 + h], hz = Hg[b * G3 + HH + h],
            hn = Hg[b * G3 + 2 * HH + h];
      float r  = sigmoidf_(xr + hr);
      float z  = sigmoidf_(xz + hz);
      float nn = tanhf(xn + r * hn);
      float hnew = (1.f - z) * nn + z * Hs[idx];
      Hs[idx] = hnew;
      gru[((size_t)b * TT + t) * (2 * HH) + (size_t)dir * HH + h] = hnew;
    }
    __syncthreads();
  }
}

// ----------------------------------------------- K4: attention + classifier
__global__ void k_head(const float* attn_W, const float* attn_b,
                       const float* clf_W, const float* clf_b,
                       const float* ws, float* out) {
  __shared__ float sc[TT];
  __shared__ float red[128];
  int b = blockIdx.x, h = threadIdx.x;  // 128 threads
  const float* gru = ws + WS_GRU + (size_t)b * TT * (2 * HH);
  for (int t = h; t < TT; t += 128) {
    float acc = attn_b[0];
    const float* gv = gru + (size_t)t * (2 * HH);
    for (int j = 0; j < 2 * HH; ++j) acc += gv[j] * attn_W[j];
    sc[t] = tanhf(acc);
  }
  __syncthreads();
  float m = -1e30f;
  for (int t = h; t < TT; t += 128) m = fmaxf(m, sc[t]);
  red[h] = m; __syncthreads();
  for (int sdv = 64; sdv > 0; sdv >>= 1) {
    if (h < sdv) red[h] = fmaxf(red[h], red[h + sdv]);
    __syncthreads();
  }
  float mm = red[0]; __syncthreads();
  float ss = 0.f;
  for (int t = h; t < TT; t += 128) { float e = __expf(sc[t] - mm); sc[t] = e; ss += e; }
  red[h] = ss; __syncthreads();
  for (int sdv = 64; sdv > 0; sdv >>= 1) {
    if (h < sdv) red[h] += red[h + sdv];
    __syncthreads();
  }
  float S = red[0]; __syncthreads();
  float ctx = 0.f;
  for (int t = 0; t < TT; ++t) ctx += sc[t] * gru[(size_t)t * (2 * HH) + h];
  ctx /= S;
  red[h] = ctx * clf_W[h]; __syncthreads();
  for (int sdv = 64; sdv > 0; sdv >>= 1) {
    if (h < sdv) red[h] += red[h + sdv];
    __syncthreads();
  }
  if (h == 0) out[b] = sigmoidf_(red[0] + clf_b[0]);
}

extern "C" void kernel_launch(void* const* d_in, const int* in_sizes, int n_in,
                              void* d_out, int out_size, void* d_ws, size_t ws_size,
                              hipStream_t stream) {
  const float* x     = (const float*)d_in[0];
  const int*   sei   = (const int*)d_in[1];
  const float* sew   = (const float*)d_in[2];
  const int*   fei   = (const int*)d_in[3];
  const float* few   = (const float*)d_in[4];
  const float* Wcheb = (const float*)d_in[5];
  const float* bcheb = (const float*)d_in[6];
  const float* Wgcn  = (const float*)d_in[7];
  const float* bgcn  = (const float*)d_in[8];
  const float* Wih_f = (const float*)d_in[9];
  const float* Whh_f = (const float*)d_in[10];
  const float* bih_f = (const float*)d_in[11];
  const float* bhh_f = (const float*)d_in[12];
  const float* Wih_b = (const float*)d_in[13];
  const float* Whh_b = (const float*)d_in[14];
  const float* bih_b = (const float*)d_in[15];
  const float* bhh_b = (const float*)d_in[16];
  const float* attn_W = (const float*)d_in[17];
  const float* attn_b = (const float*)d_in[18];
  const float* clf_W  = (const float*)d_in[19];
  const float* clf_b  = (const float*)d_in[20];
  float* ws  = (float*)d_ws;
  float* out = (float*)d_out;

  k_build_graphs<<<1, 256, 0, stream>>>(sei, sew, fei, few, ws);
  k_fuse_weights<<<GALL, 64, 0, stream>>>(Wcheb, bcheb, Wgcn, bgcn,
                                          Wih_f, bih_f, Wih_b, bih_b, ws);
  k_xp_gemm<<<dim3(6, BB), 256, 0, stream>>>(x, ws);
  k_gru<<<2, 512, 28672 * sizeof(float), stream>>>(Whh_f, bhh_f, Whh_b, bhh_b, ws);
  k_head<<<BB, 128, 0, stream>>>(attn_W, attn_b, clf_W, clf_b, ws, out);
}